// GraphormerGraphAttnBias_55327768707635
// MI455X (gfx1250) — compile-verified
//
#include <hip/hip_runtime.h>
#include <hip/hip_bf16.h>

// ---------------- problem constants ----------------
#define BB   4
#define NN   256
#define HH   32
#define MHD  5
#define FF   3
#define NP1  257

// ---------------- tiling ----------------
#define WAVES_PER_WG    8
#define PAIRS_PER_WAVE  16
#define PAIRS_PER_WG    128          // 8 waves * 16 pairs
#define KCAT            160          // 5 * 32 concatenated K
#define NCHUNK          40           // KCAT / 4
#define A_STRIDE        164          // padded row stride (floats) for A staging
#define D_STRIDE        33           // padded row stride for D staging

// ---------------- LDS layout (floats) ----------------
// W is stored k-pair interleaved: Wr[t*64 + n*2 + p] = W[2t+p][n],  t=0..79, n=0..31, p=0..1
#define OFF_W      0                                 // 5120 : interleaved W_cat
#define OFF_A      (OFF_W + KCAT*HH)                 // 20992: A [8 waves][16][A_STRIDE]
#define OFF_IDX    (OFF_A + WAVES_PER_WG*PAIRS_PER_WAVE*A_STRIDE)  // 1920 ints
#define OFF_D      (OFF_IDX + WAVES_PER_WG*240)      // 4224 : D [128][33]
#define OFF_SPOS   (OFF_D + PAIRS_PER_WG*D_STRIDE)   // 128 ints
#define OFF_INVSP  (OFF_SPOS + PAIRS_PER_WG)         // 128 floats
#define OFF_AB     (OFF_INVSP + PAIRS_PER_WG)        // 128 floats (2*attn_bias row)
#define SMEM_FLOATS (OFF_AB + PAIRS_PER_WG)          // total = 32640 floats (~127.5 KB)

typedef __attribute__((ext_vector_type(2))) float v2f;
typedef __attribute__((ext_vector_type(8))) float v8f;

// ============================================================
// Main kernel: interior gab[:, :, 1:, 1:]
// One WG = (b, i, 128 consecutive j). One wave = 16 pairs.
// ============================================================
__global__ __launch_bounds__(256) void graphormer_main(
    const float* __restrict__ attn_bias,      // (B, 257, 257)
    const int*   __restrict__ spatial_pos,    // (B, 256, 256)
    const int*   __restrict__ input_edges,    // (B, 256, 256, 5, 3)
    const float* __restrict__ edge_encoder_w, // (1537, 32)   L2-resident (192 KB)
    const float* __restrict__ spatial_pos_w,  // (512, 32)
    const float* __restrict__ edge_dis_w,     // (128*32*32, 1): first 5120 = W_cat[160][32]
    float*       __restrict__ out)            // (B, 32, 257, 257)
{
    extern __shared__ float smem[];
    float* Wlds  = smem + OFF_W;
    float* Alds  = smem + OFF_A;
    int*   IDX   = (int*)(smem + OFF_IDX);
    float* Dlds  = smem + OFF_D;
    int*   SPOS  = (int*)(smem + OFF_SPOS);
    float* INVSP = smem + OFF_INVSP;
    float* ABR   = smem + OFF_AB;

    const int tid  = threadIdx.x;
    const int lane = tid & 31;
    const int wave = tid >> 5;

    const int blk = blockIdx.x;              // 0 .. 2047
    const int b   = blk >> 9;                // / 512
    const int i   = (blk >> 1) & 255;
    const int j0  = (blk & 1) * PAIRS_PER_WG;

    // ---- stage W_cat into LDS, k-pair interleaved:
    //      Wr[(k>>1)*64 + n*2 + (k&1)] = W[k][n]
    #pragma unroll
    for (int r = 0; r < (KCAT * HH) / 256; ++r) {
        const int e = r * 256 + tid;         // flat index into W_cat [k][n]
        const int k = e >> 5;
        const int n = e & 31;
        Wlds[(k >> 1) * 64 + n * 2 + (k & 1)] = edge_dis_w[e];
    }

    // ---- coalesced load of this wave's 240 edge indices ----
    const int jw = j0 + wave * PAIRS_PER_WAVE;
    const long long pairBase =
        (((long long)b * NN + i) * NN + jw) * (long long)(MHD * FF);
    int* idxw = IDX + wave * 240;
    for (int e = lane; e < PAIRS_PER_WAVE * MHD * FF; e += 32)
        idxw[e] = input_edges[pairBase + e];

    // ---- per-pair spatial-pos data + 2*attn_bias row ----
    if (tid < PAIRS_PER_WG) {
        const int j = j0 + tid;
        const int sp = spatial_pos[(((long long)b * NN) + i) * NN + j];
        SPOS[tid] = sp;
        int s = (sp == 0) ? 1 : sp;
        s = (s > 1) ? (s - 1) : s;
        s = (s > MHD) ? MHD : s;
        INVSP[tid] = 1.0f / (float)s;
        ABR[tid] = 2.0f * attn_bias[(((long long)b * NP1) + (1 + i)) * NP1 + (1 + j)];
    }
    __syncthreads();

    // ---- gather + mean into A staging: Aw[p][d*32 + h_in], lane = h_in ----
    float* Aw = Alds + wave * PAIRS_PER_WAVE * A_STRIDE;
    const float inv3 = 1.0f / 3.0f;
    #pragma unroll
    for (int p = 0; p < PAIRS_PER_WAVE; ++p) {
        #pragma unroll
        for (int d = 0; d < MHD; ++d) {
            const int e0 = idxw[p * 15 + d * 3 + 0];
            const int e1 = idxw[p * 15 + d * 3 + 1];
            const int e2 = idxw[p * 15 + d * 3 + 2];
            const float acc = edge_encoder_w[e0 * HH + lane]
                            + edge_encoder_w[e1 * HH + lane]
                            + edge_encoder_w[e2 * HH + lane];
            Aw[p * A_STRIDE + d * HH + lane] = acc * inv3;
        }
    }
    __syncthreads();   // A (wave-local) + W visible to compute

    // ---- WMMA: D[16x32] = A[16x160] * W_cat[160x32], K split into 40 chunks ----
    // A frag (16x4 f32): lanes 0-15 -> M=lane, K={4c,4c+1}; lanes 16-31 -> M=lane-16, K={4c+2,4c+3}
    // B frag (4x16 f32): vgpr0 -> row K=4c (lo half) / 4c+2 (hi half); vgpr1 -> +1; N = lane&15.
    //   With interleaved Wr, each lane's (b.x, b.y) is one aligned ds_load_b64.
    v8f c0 = {0.f,0.f,0.f,0.f,0.f,0.f,0.f,0.f};
    v8f c1 = {0.f,0.f,0.f,0.f,0.f,0.f,0.f,0.f};
    const int m    = lane & 15;
    const int koff = (lane < 16) ? 0 : 2;     // K sub-offset for this half-wave
    const float* Arow = Aw + m * A_STRIDE + koff;
    const int nA = lane & 15;
    // lane's W base: t = 2c + koff/2  ->  Wr + t*64 + n*2
    const float* Wrow = Wlds + (koff >> 1) * 64 + nA * 2;

    for (int c = 0; c < NCHUNK; ++c) {
        const v2f a  = *(const v2f*)(Arow + 4 * c);            // ds_load_b64
        const v2f b0 = *(const v2f*)(Wrow + 128 * c);          // ds_load_b64
        const v2f b1 = *(const v2f*)(Wrow + 128 * c + 32);     // ds_load_b64
        c0 = __builtin_amdgcn_wmma_f32_16x16x4_f32(false, a, false, b0,
                                                   (short)0, c0, false, false);
        c1 = __builtin_amdgcn_wmma_f32_16x16x4_f32(false, a, false, b1,
                                                   (short)0, c1, false, false);
    }

    // ---- write D fragments to LDS (C/D layout: vgpr r -> M=r (lo half) / r+8 (hi half)) ----
    const int mbase = (lane < 16) ? 0 : 8;
    const int nD = lane & 15;
    #pragma unroll
    for (int r = 0; r < 8; ++r) {
        const int row = wave * PAIRS_PER_WAVE + mbase + r;
        Dlds[row * D_STRIDE + nD]      = c0[r];
        Dlds[row * D_STRIDE + 16 + nD] = c1[r];
    }
    __syncthreads();

    // ---- epilogue: out = D/sp + spatial_pos_w[spos][h] + 2*attn_bias, coalesced ----
    #pragma unroll
    for (int rep = 0; rep < 16; ++rep) {
        const int e  = rep * 256 + tid;      // 0 .. 4095
        const int h  = e >> 7;               // 0 .. 31
        const int pk = e & 127;              // 0 .. 127
        const float v = Dlds[pk * D_STRIDE + h] * INVSP[pk]
                      + spatial_pos_w[SPOS[pk] * HH + h]
                      + ABR[pk];
        out[((((long long)b * HH) + h) * NP1 + (1 + i)) * NP1 + (1 + j0 + pk)] = v;
    }
}

// ============================================================
// Border kernel: row 0 and column 0 (virtual node)
// ============================================================
__global__ __launch_bounds__(256) void graphormer_border(
    const float* __restrict__ attn_bias,
    const float* __restrict__ virt_w,
    float*       __restrict__ out)
{
    const int idx = blockIdx.x * blockDim.x + threadIdx.x;
    const int total = BB * HH * (NP1 + NN);      // 257 row-0 + 256 col-0 entries
    if (idx >= total) return;
    const int bh = idx / (NP1 + NN);
    const int e  = idx % (NP1 + NN);
    const int b  = bh / HH;
    const int h  = bh % HH;
    const float t = virt_w[h];
    if (e < NP1) {
        // gab[b,h,0,k] = 2*attn_bias[b,0,k] + virt_w[h]
        out[((((long long)b * HH) + h) * NP1 + 0) * NP1 + e] =
            2.0f * attn_bias[((long long)b * NP1 + 0) * NP1 + e] + t;
    } else {
        // gab[b,h,q,0] = 2*attn_bias[b,q,0] + virt_w[h],  q = 1..256
        const int q = e - NN;                     // 1 .. 256
        out[((((long long)b * HH) + h) * NP1 + q) * NP1 + 0] =
            2.0f * attn_bias[((long long)b * NP1 + q) * NP1 + 0] + t;
    }
}

// ============================================================
extern "C" void kernel_launch(void* const* d_in, const int* in_sizes, int n_in,
                              void* d_out, int out_size, void* d_ws, size_t ws_size,
                              hipStream_t stream) {
    (void)in_sizes; (void)n_in; (void)out_size; (void)d_ws; (void)ws_size;
    // setup_inputs order:
    // 0 input_nodes(int) 1 attn_bias(f32) 2 spatial_pos(int) 3 input_edges(int)
    // 4 attn_edge_type(int, unused) 5 edge_encoder_w(f32) 6 spatial_pos_w(f32)
    // 7 edge_dis_w(f32) 8 virt_w(f32)
    const float* attn_bias      = (const float*)d_in[1];
    const int*   spatial_pos    = (const int*)d_in[2];
    const int*   input_edges    = (const int*)d_in[3];
    const float* edge_encoder_w = (const float*)d_in[5];
    const float* spatial_pos_w  = (const float*)d_in[6];
    const float* edge_dis_w     = (const float*)d_in[7];
    const float* virt_w         = (const float*)d_in[8];
    float*       out            = (float*)d_out;

    const size_t smem_bytes = (size_t)SMEM_FLOATS * sizeof(float);
    const int grid_main = BB * NN * (NN / PAIRS_PER_WG);   // 2048
    graphormer_main<<<grid_main, 256, smem_bytes, stream>>>(
        attn_bias, spatial_pos, input_edges,
        edge_encoder_w, spatial_pos_w, edge_dis_w, out);

    const int total_border = BB * HH * (NP1 + NN);
    graphormer_border<<<(total_border + 255) / 256, 256, 0, stream>>>(
        attn_bias, virt_w, out);
}